// GNNModel_66322884985041
// MI455X (gfx1250) — compile-verified
//
#include <hip/hip_runtime.h>
#include <hip/hip_bf16.h>
#include <math.h>

// ---------------------------------------------------------------------------
// 2-layer GCN for MI455X (gfx1250, wave32).
// GEMMs use V_WMMA_F32_16X16X4_F32 (exact fp32 tensor path, CDNA5-only).
// Edge aggregation is float4-vectorized gather + global_atomic_add_f32 scatter.
// ---------------------------------------------------------------------------

#define N_NODES 50000
#define N_EDGES 800000
#define IN_DIM  128
#define HID_DIM 64   // == OUT_DIM; all per-node feature rows after layer 1 are 64 wide

typedef float v2f __attribute__((ext_vector_type(2)));
typedef float v8f __attribute__((ext_vector_type(8)));

// ---------------- degree / norm ----------------

__global__ void init_deg_kernel(float* deg, int n) {
    int i = blockIdx.x * blockDim.x + threadIdx.x;
    if (i < n) deg[i] = 1.0f;  // self-loop contributes 1 to in-degree
}

__global__ void edge_deg_kernel(const int* __restrict__ dst, float* deg, int e_count) {
    int e = blockIdx.x * blockDim.x + threadIdx.x;
    if (e < e_count) atomicAdd(&deg[dst[e]], 1.0f);
}

__global__ void rsqrt_kernel(float* deg, int n) {
    int i = blockIdx.x * blockDim.x + threadIdx.x;
    if (i < n) {
        float d = deg[i];
        deg[i] = (d > 0.0f) ? rsqrtf(d) : 0.0f;  // in-place: deg -> dinv
    }
}

// ---------------- fp32 WMMA GEMM:  C[M,64] = A[M,K] * W[K,64] ----------------
// Block = 256 threads = 8 waves. Wave w -> output tile (m_tile = blk*2 + w/4,
// n_tile = w%4). W is staged in LDS (K*64 floats). K stepped by 4 per WMMA.
//
// A fragment (16x4 f32, ISA 7.12.2): lanes 0-15 hold {K=0,K=1} of row M=lane,
// lanes 16-31 hold {K=2,K=3} of row M=lane-16.
// B fragment (4x16): row-striped across lanes within a VGPR: lanes 0-15 get
// {K=0,K=1} at N=lane, lanes 16-31 get {K=2,K=3} at N=lane-16.
// C/D (16x16 f32): VGPR j: lanes 0-15 -> M=j, N=lane; lanes 16-31 -> M=j+8.

template <int K_DIM>
__global__ __launch_bounds__(256) void gemm_wmma_f32(const float* __restrict__ A,
                                                     const float* __restrict__ W,
                                                     float* __restrict__ C, int M) {
    __shared__ float sW[K_DIM * 64];
    for (int i = threadIdx.x; i < K_DIM * 64; i += 256) sW[i] = W[i];
    __syncthreads();

    const int wave = threadIdx.x >> 5;
    const int lane = threadIdx.x & 31;
    const int m_tile = blockIdx.x * 2 + (wave >> 2);
    const int n_tile = wave & 3;

    const int row16 = lane & 15;           // matrix row within tile (A) / col (B)
    const int khalf = (lane >> 4) << 1;    // 0 for lanes 0-15, 2 for lanes 16-31

    int mrow = m_tile * 16 + row16;
    int arow = (mrow < M) ? mrow : (M - 1);            // clamp; masked on store
    const float* __restrict__ Aptr = A + (size_t)arow * K_DIM;
    const float* __restrict__ Bcol = sW + n_tile * 16 + row16;

    v8f acc = {};
    #pragma unroll
    for (int k = 0; k < K_DIM; k += 4) {
        v2f a, b;
        a.x = Aptr[k + khalf];
        a.y = Aptr[k + khalf + 1];
        b.x = Bcol[(k + khalf) * 64];
        b.y = Bcol[(k + khalf + 1) * 64];
        acc = __builtin_amdgcn_wmma_f32_16x16x4_f32(
            /*neg_a=*/false, a, /*neg_b=*/false, b,
            /*c_mod=*/(short)0, acc, /*reuse_a=*/false, /*reuse_b=*/false);
    }

    const int mbase = m_tile * 16 + ((lane >> 4) << 3);  // +8 for upper half-wave
    const int ncol  = n_tile * 16 + row16;
    #pragma unroll
    for (int j = 0; j < 8; ++j) {
        int m = mbase + j;
        if (m < M) C[(size_t)m * 64 + ncol] = acc[j];
    }
}

// ---------------- zero workspace ----------------

__global__ void zero_kernel(float4* p, int n4) {
    int i = blockIdx.x * blockDim.x + threadIdx.x;
    if (i < n4) p[i] = make_float4(0.f, 0.f, 0.f, 0.f);
}

// ---------------- edge scatter:  agg[dst] += h[src] * dinv[src]*dinv[dst] ----

__global__ __launch_bounds__(256) void scatter_kernel(const float* __restrict__ h,
                                                      const float* __restrict__ dinv,
                                                      const int* __restrict__ src,
                                                      const int* __restrict__ dst,
                                                      float* __restrict__ agg, int e_count) {
    int gid = blockIdx.x * blockDim.x + threadIdx.x;
    int e = gid >> 4;                  // 16 threads per edge (64 floats / float4)
    if (e >= e_count) return;
    int c = (gid & 15) << 2;
    int s = src[e], d = dst[e];
    float norm = dinv[s] * dinv[d];
    const float4 hv = *(const float4*)(h + (size_t)s * 64 + c);
    float* o = agg + (size_t)d * 64 + c;
    atomicAdd(o + 0, hv.x * norm);
    atomicAdd(o + 1, hv.y * norm);
    atomicAdd(o + 2, hv.z * norm);
    atomicAdd(o + 3, hv.w * norm);
}

// -------- finalize: out = [relu](agg + h*dinv^2 (self-loop) + bias) ---------

__global__ __launch_bounds__(256) void finalize_kernel(const float* __restrict__ agg,
                                                       const float* __restrict__ h,
                                                       const float* __restrict__ dinv,
                                                       const float* __restrict__ bias,
                                                       float* __restrict__ out,
                                                       int n_nodes, int do_relu) {
    int gid = blockIdx.x * blockDim.x + threadIdx.x;
    int n = gid >> 4;
    if (n >= n_nodes) return;
    int c = (gid & 15) << 2;
    float di = dinv[n];
    float self_n = di * di;
    const float4 a  = *(const float4*)(agg + (size_t)n * 64 + c);
    const float4 hv = *(const float4*)(h   + (size_t)n * 64 + c);
    const float4 b  = *(const float4*)(bias + c);
    float4 r;
    r.x = a.x + hv.x * self_n + b.x;
    r.y = a.y + hv.y * self_n + b.y;
    r.z = a.z + hv.z * self_n + b.z;
    r.w = a.w + hv.w * self_n + b.w;
    if (do_relu) {
        r.x = fmaxf(r.x, 0.f); r.y = fmaxf(r.y, 0.f);
        r.z = fmaxf(r.z, 0.f); r.w = fmaxf(r.w, 0.f);
    }
    *(float4*)(out + (size_t)n * 64 + c) = r;
}

// ---------------------------------------------------------------------------

extern "C" void kernel_launch(void* const* d_in, const int* in_sizes, int n_in,
                              void* d_out, int out_size, void* d_ws, size_t ws_size,
                              hipStream_t stream) {
    const float* x  = (const float*)d_in[0];             // [50000,128]
    const int*   ei = (const int*)d_in[1];               // [2,800000] (JAX x64-off -> int32)
    const float* W1 = (const float*)d_in[2];             // [128,64]
    const float* b1 = (const float*)d_in[3];             // [64]
    const float* W2 = (const float*)d_in[4];             // [64,64]
    const float* b2 = (const float*)d_in[5];             // [64]
    float* out = (float*)d_out;                          // [50000,64]

    const int* src = ei;                                 // edge_index[0]
    const int* dst = ei + N_EDGES;                       // edge_index[1]

    // workspace layout (floats): dinv[64K slot] | buf0[50000*64] | buf1[50000*64]
    float* dinv = (float*)d_ws;
    float* buf0 = dinv + 65536;
    float* buf1 = buf0 + (size_t)N_NODES * 64;
    const int feat_elems = N_NODES * 64;                 // 3,200,000

    const int B = 256;
    dim3 blk(B);

    // --- degree / symmetric norm ---
    init_deg_kernel<<<(N_NODES + B - 1) / B, blk, 0, stream>>>(dinv, N_NODES);
    edge_deg_kernel<<<(N_EDGES + B - 1) / B, blk, 0, stream>>>(dst, dinv, N_EDGES);
    rsqrt_kernel<<<(N_NODES + B - 1) / B, blk, 0, stream>>>(dinv, N_NODES);

    // --- layer 1 ---
    int gemm_grid = (N_NODES + 31) / 32;                 // 32 rows per block
    gemm_wmma_f32<IN_DIM><<<gemm_grid, blk, 0, stream>>>(x, W1, buf0, N_NODES);     // H1 -> buf0
    zero_kernel<<<(feat_elems / 4 + B - 1) / B, blk, 0, stream>>>((float4*)buf1, feat_elems / 4);
    scatter_kernel<<<(N_EDGES * 16 + B - 1) / B, blk, 0, stream>>>(buf0, dinv, src, dst, buf1, N_EDGES);
    finalize_kernel<<<(N_NODES * 16 + B - 1) / B, blk, 0, stream>>>(buf1, buf0, dinv, b1,
                                                                    buf0, N_NODES, /*relu=*/1);

    // --- layer 2 ---
    gemm_wmma_f32<HID_DIM><<<gemm_grid, blk, 0, stream>>>(buf0, W2, buf1, N_NODES); // H2 -> buf1
    zero_kernel<<<(feat_elems / 4 + B - 1) / B, blk, 0, stream>>>((float4*)buf0, feat_elems / 4);
    scatter_kernel<<<(N_EDGES * 16 + B - 1) / B, blk, 0, stream>>>(buf1, dinv, src, dst, buf0, N_EDGES);
    finalize_kernel<<<(N_NODES * 16 + B - 1) / B, blk, 0, stream>>>(buf0, buf1, dinv, b2,
                                                                    out, N_NODES, /*relu=*/0);
}